// GraphEncoder_66194035966394
// MI455X (gfx1250) — compile-verified
//
#include <hip/hip_runtime.h>
#include <hip/hip_bf16.h>

typedef __attribute__((ext_vector_type(2))) float v2f;
typedef __attribute__((ext_vector_type(8))) float v8f;

#define NEG_SLOPE 0.2f
#define SOFT_EPS  1e-16f
// encoded -inf for monotonic-uint float max
#define ENC_NEG_INF 0x007FFFFFu

__device__ __forceinline__ unsigned enc_f32(float x) {
    unsigned b = __float_as_uint(x);
    return (b & 0x80000000u) ? ~b : (b | 0x80000000u);
}
__device__ __forceinline__ float dec_f32(unsigned k) {
    unsigned b = (k & 0x80000000u) ? (k ^ 0x80000000u) : ~k;
    return __uint_as_float(b);
}

// ---------------------------------------------------------------------------
// C[M x NCOLS] = A[M x K] * W[NCOLS x K]^T  (row-major), f32 WMMA.
// One wave computes a 16x64 strip: 4 accumulators share each A fragment,
// so A-fragment traffic is amortized 4x (intensity ~6.5 FLOP/B vs ~2).
// ---------------------------------------------------------------------------
template <int K, int NCOLS>
__global__ void wmma_gemm_nt4(const float* __restrict__ A,
                              const float* __restrict__ W,
                              float* __restrict__ C, int M) {
    static_assert(NCOLS % 64 == 0, "NCOLS must be a multiple of 64");
    const int lane     = threadIdx.x & 31;
    const int wave     = threadIdx.x >> 5;
    const int groups_n = NCOLS / 64;
    const int grp  = blockIdx.x * (blockDim.x >> 5) + wave;
    const int gm   = grp / groups_n;
    const int gn   = grp % groups_n;
    if (gm * 16 >= M) return;            // wave-uniform guard (EXEC all-1s)

    // A fragment (16x4 f32): lanes 0-15 -> M=lane, K={k0,k0+1}; lanes 16-31 -> K={k0+2,k0+3}
    const int mrow  = gm * 16 + (lane & 15);
    const int khalf = (lane >> 4) * 2;
    const float* ap  = A + (size_t)mrow * K + khalf;
    // B fragments: 4 tiles at W rows gn*64 + {0,16,32,48} + (lane&15)
    const float* wp0 = W + (size_t)(gn * 64 + (lane & 15)) * K + khalf;

    v8f c0 = {0.f,0.f,0.f,0.f,0.f,0.f,0.f,0.f};
    v8f c1 = c0, c2 = c0, c3 = c0;

#pragma unroll 4
    for (int k0 = 0; k0 < K; k0 += 4) {
        v2f a, b0, b1, b2, b3;
        a.x  = ap[k0];            a.y  = ap[k0 + 1];
        b0.x = wp0[k0];           b0.y = wp0[k0 + 1];
        b1.x = wp0[k0 + 16 * K];  b1.y = wp0[k0 + 16 * K + 1];
        b2.x = wp0[k0 + 32 * K];  b2.y = wp0[k0 + 32 * K + 1];
        b3.x = wp0[k0 + 48 * K];  b3.y = wp0[k0 + 48 * K + 1];
        c0 = __builtin_amdgcn_wmma_f32_16x16x4_f32(false, a, false, b0, (short)0, c0, false, false);
        c1 = __builtin_amdgcn_wmma_f32_16x16x4_f32(false, a, false, b1, (short)0, c1, false, false);
        c2 = __builtin_amdgcn_wmma_f32_16x16x4_f32(false, a, false, b2, (short)0, c2, false, false);
        c3 = __builtin_amdgcn_wmma_f32_16x16x4_f32(false, a, false, b3, (short)0, c3, false, false);
    }

    // C/D layout: VGPR r -> row (r + 8*(lane>=16)), col = lane&15
    const int row0 = gm * 16 + ((lane >> 4) << 3);
    const int col0 = gn * 64 + (lane & 15);
    float* cp = C + (size_t)row0 * NCOLS + col0;
#pragma unroll
    for (int r = 0; r < 8; ++r) {
        cp[(size_t)r * NCOLS +  0] = c0[r];
        cp[(size_t)r * NCOLS + 16] = c1[r];
        cp[(size_t)r * NCOLS + 32] = c2[r];
        cp[(size_t)r * NCOLS + 48] = c3[r];
    }
}

// ---------------------------------------------------------------------------
// alpha_s[n,h] = sum_d h[n,h,d]*a_src[h,d] ; alpha_d likewise. One wave/node.
// ---------------------------------------------------------------------------
template <int H, int D>
__global__ void alpha_kernel(const float* __restrict__ hmat,
                             const float* __restrict__ a_src,
                             const float* __restrict__ a_dst,
                             float* __restrict__ out_s,
                             float* __restrict__ out_d, int N) {
    const int lane = threadIdx.x & 31;
    const int node = blockIdx.x * (blockDim.x >> 5) + (threadIdx.x >> 5);
    if (node >= N) return;
    const float* hp = hmat + (size_t)node * H * D;

    float accs[H], accd[H];
#pragma unroll
    for (int i = 0; i < H; ++i) { accs[i] = 0.f; accd[i] = 0.f; }

#pragma unroll
    for (int t = 0; t < (H * D) / 32; ++t) {
        const int j = t * 32 + lane;
        const int hh = (t * 32) / D;         // uniform per t since 32 | D
        const float v = hp[j];
        accs[hh] += v * a_src[j];
        accd[hh] += v * a_dst[j];
    }
#pragma unroll
    for (int hh = 0; hh < H; ++hh) {
        float vs = accs[hh], vd = accd[hh];
        for (int off = 16; off > 0; off >>= 1) {
            vs += __shfl_xor(vs, off, 32);
            vd += __shfl_xor(vd, off, 32);
        }
        if (lane == 0) {
            out_s[node * H + hh] = vs;
            out_d[node * H + hh] = vd;
        }
    }
}

// ---------------------------------------------------------------------------
// Edge pass 1: e = leaky_relu(as[src]+ad[dst]); segment-max into mkey[dst]
// ---------------------------------------------------------------------------
template <int H>
__global__ void edge_pass1(const int* __restrict__ ei, int E, int N,
                           const float* __restrict__ as,
                           const float* __restrict__ ad,
                           float* __restrict__ ebuf,
                           unsigned* __restrict__ mkey) {
    const long long idx = blockIdx.x * (long long)blockDim.x + threadIdx.x;
    const long long total = (long long)(E + N) * H;
    if (idx >= total) return;
    const int edge = (int)(idx / H);
    const int head = (int)(idx % H);
    const int s = (edge < E) ? ei[edge] : edge - E;
    const int d = (edge < E) ? ei[E + edge] : edge - E;
    float e = as[s * H + head] + ad[d * H + head];
    e = (e > 0.f) ? e : NEG_SLOPE * e;
    ebuf[idx] = e;
    atomicMax(&mkey[d * H + head], enc_f32(e));
}

// ---------------------------------------------------------------------------
// Edge pass 2: ex = exp(e - m[dst]); segment-sum into ssum[dst]
// ---------------------------------------------------------------------------
template <int H>
__global__ void edge_pass2(const int* __restrict__ ei, int E, int N,
                           const unsigned* __restrict__ mkey,
                           float* __restrict__ ebuf,
                           float* __restrict__ ssum) {
    const long long idx = blockIdx.x * (long long)blockDim.x + threadIdx.x;
    const long long total = (long long)(E + N) * H;
    if (idx >= total) return;
    const int edge = (int)(idx / H);
    const int head = (int)(idx % H);
    const int d = (edge < E) ? ei[E + edge] : edge - E;
    const float m = dec_f32(mkey[d * H + head]);
    const float ex = __expf(ebuf[idx] - m);
    ebuf[idx] = ex;
    atomicAdd(&ssum[d * H + head], ex);
}

// ---------------------------------------------------------------------------
// Edge pass 3: acc[dst, r] += h[src, r] * ex/(ssum[dst]+eps). 1 thread/(edge,r);
// consecutive threads cover consecutive channels -> coalesced loads & atomics.
// ---------------------------------------------------------------------------
template <int H, int D>
__global__ void edge_pass3(const int* __restrict__ ei, int E, int N,
                           const float* __restrict__ hmat,
                           const float* __restrict__ ebuf,
                           const float* __restrict__ ssum,
                           float* __restrict__ acc) {
    const long long idx = blockIdx.x * (long long)blockDim.x + threadIdx.x;
    const long long total = (long long)(E + N) * H * D;
    if (idx >= total) return;
    const int edge = (int)(idx / (H * D));
    const int r    = (int)(idx % (H * D));
    const int head = r / D;
    const int s = (edge < E) ? ei[edge] : edge - E;
    const int d = (edge < E) ? ei[E + edge] : edge - E;
    const float alpha = ebuf[(long long)edge * H + head] /
                        (ssum[d * H + head] + SOFT_EPS);
    atomicAdd(&acc[(size_t)d * H * D + r], hmat[(size_t)s * H * D + r] * alpha);
}

// ---------------------------------------------------------------------------
// z = elu(z + bias[j])  (in place)
// ---------------------------------------------------------------------------
__global__ void bias_elu(float* __restrict__ z, const float* __restrict__ b,
                         int HD, long long total) {
    const long long idx = blockIdx.x * (long long)blockDim.x + threadIdx.x;
    if (idx >= total) return;
    const float v = z[idx] + b[(int)(idx % HD)];
    z[idx] = (v > 0.f) ? v : (__expf(v) - 1.f);
}

__global__ void fill_u32(unsigned* p, unsigned v, long long n) {
    const long long i = blockIdx.x * (long long)blockDim.x + threadIdx.x;
    if (i < n) p[i] = v;
}
__global__ void fill_f32(float* p, float v, long long n) {
    const long long i = blockIdx.x * (long long)blockDim.x + threadIdx.x;
    if (i < n) p[i] = v;
}
__global__ void seed_bias(float* __restrict__ out, const float* __restrict__ b,
                          int D, long long n) {
    const long long i = blockIdx.x * (long long)blockDim.x + threadIdx.x;
    if (i < n) out[i] = b[(int)(i % D)];
}

static inline unsigned gridFor(long long total, int block) {
    return (unsigned)((total + block - 1) / block);
}

extern "C" void kernel_launch(void* const* d_in, const int* in_sizes, int n_in,
                              void* d_out, int out_size, void* d_ws, size_t ws_size,
                              hipStream_t stream) {
    const float* x      = (const float*)d_in[0];
    const int*   ei     = (const int*)d_in[1];
    const float* W1     = (const float*)d_in[2];
    const float* a_src1 = (const float*)d_in[3];
    const float* a_dst1 = (const float*)d_in[4];
    const float* b1     = (const float*)d_in[5];
    const float* W2     = (const float*)d_in[6];
    const float* a_src2 = (const float*)d_in[7];
    const float* a_dst2 = (const float*)d_in[8];
    const float* b2     = (const float*)d_in[9];
    float* out = (float*)d_out;

    const int N = in_sizes[0] / 256;   // 50000
    const int E = in_sizes[1] / 2;     // 800000
    const int EN = E + N;

    // workspace carve-up (all 4-byte typed)
    char* ws = (char*)d_ws;
    float*    h1   = (float*)ws;                   ws += (size_t)N * 256 * 4;
    float*    acc1 = (float*)ws;                   ws += (size_t)N * 256 * 4;
    float*    as1  = (float*)ws;                   ws += (size_t)N * 4 * 4;
    float*    ad1  = (float*)ws;                   ws += (size_t)N * 4 * 4;
    unsigned* mk1  = (unsigned*)ws;                ws += (size_t)N * 4 * 4;
    float*    s1   = (float*)ws;                   ws += (size_t)N * 4 * 4;
    float*    e1   = (float*)ws;                   ws += (size_t)EN * 4 * 4;
    float*    h2   = (float*)ws;                   ws += (size_t)N * 64 * 4;
    float*    as2  = (float*)ws;                   ws += (size_t)N * 4;
    float*    ad2  = (float*)ws;                   ws += (size_t)N * 4;
    unsigned* mk2  = (unsigned*)ws;                ws += (size_t)N * 4;
    float*    s2   = (float*)ws;                   ws += (size_t)N * 4;
    float*    e2   = (float*)ws;                   ws += (size_t)EN * 4;

    const int B = 256;

    // init scratch + seed output with bias2
    fill_u32<<<gridFor((long long)N * 4, B), B, 0, stream>>>(mk1, ENC_NEG_INF, (long long)N * 4);
    fill_f32<<<gridFor((long long)N * 4, B), B, 0, stream>>>(s1, 0.f, (long long)N * 4);
    fill_f32<<<gridFor((long long)N * 256, B), B, 0, stream>>>(acc1, 0.f, (long long)N * 256);
    fill_u32<<<gridFor(N, B), B, 0, stream>>>(mk2, ENC_NEG_INF, N);
    fill_f32<<<gridFor(N, B), B, 0, stream>>>(s2, 0.f, N);
    seed_bias<<<gridFor((long long)N * 64, B), B, 0, stream>>>(out, b2, 64, (long long)N * 64);

    // ---- Layer 1 ----
    {
        const long long groups = (long long)((N + 15) / 16) * (256 / 64);
        wmma_gemm_nt4<256, 256><<<gridFor(groups, 8), B, 0, stream>>>(x, W1, h1, N);
    }
    alpha_kernel<4, 64><<<gridFor(N, 8), B, 0, stream>>>(h1, a_src1, a_dst1, as1, ad1, N);
    edge_pass1<4><<<gridFor((long long)EN * 4, B), B, 0, stream>>>(ei, E, N, as1, ad1, e1, mk1);
    edge_pass2<4><<<gridFor((long long)EN * 4, B), B, 0, stream>>>(ei, E, N, mk1, e1, s1);
    edge_pass3<4, 64><<<gridFor((long long)EN * 256, B), B, 0, stream>>>(ei, E, N, h1, e1, s1, acc1);
    bias_elu<<<gridFor((long long)N * 256, B), B, 0, stream>>>(acc1, b1, 256, (long long)N * 256);

    // ---- Layer 2 ----
    {
        const long long groups = (long long)((N + 15) / 16) * (64 / 64);
        wmma_gemm_nt4<256, 64><<<gridFor(groups, 8), B, 0, stream>>>(acc1, W2, h2, N);
    }
    alpha_kernel<1, 64><<<gridFor(N, 8), B, 0, stream>>>(h2, a_src2, a_dst2, as2, ad2, N);
    edge_pass1<1><<<gridFor((long long)EN, B), B, 0, stream>>>(ei, E, N, as2, ad2, e2, mk2);
    edge_pass2<1><<<gridFor((long long)EN, B), B, 0, stream>>>(ei, E, N, mk2, e2, s2);
    edge_pass3<1, 64><<<gridFor((long long)EN * 64, B), B, 0, stream>>>(ei, E, N, h2, e2, s2, out);
}